// Phase_Feature_Generation_55525337203313
// MI455X (gfx1250) — compile-verified
//
#include <hip/hip_runtime.h>
#include <math.h>

typedef __attribute__((ext_vector_type(2))) float v2f;
typedef __attribute__((ext_vector_type(8))) float v8f;
typedef __attribute__((ext_vector_type(4))) int   v4i;

#define B_  8
#define T_  1024
#define P_  7
#define D_  512
#define CH  16          // timesteps per WMMA prefix chunk
#define NC  (T_ / CH)   // 64 chunks
#define NTILE (D_ / 16) // 32 d-tiles of width 16
#define DPAD 4          // LDS row pad (floats): row stride 516 dwords, 16B aligned

// ---------------------------------------------------------------------------
// gfx1250 async global->LDS copy of one 16B packet (ASYNCcnt-tracked DMA).
// Builtin signature (from hipcc diagnostic): param0 = addrspace(1) int4*,
// i.e. (global src, lds dst, imm offset, imm cpol).
// ---------------------------------------------------------------------------
typedef __attribute__((address_space(1))) v4i gv4i;
typedef __attribute__((address_space(3))) v4i lv4i;

__device__ __forceinline__ void async_copy_b128(const float* gsrc, float* ldst) {
#if __has_builtin(__builtin_amdgcn_global_load_async_to_lds_b128)
    __builtin_amdgcn_global_load_async_to_lds_b128((gv4i*)gsrc, (lv4i*)ldst, 0, 0);
#else
    const unsigned loff =
        (unsigned)(unsigned long long)(__attribute__((address_space(3))) char*)ldst;
    asm volatile("global_load_async_to_lds_b128 %0, %1, off"
                 :: "v"(loff), "v"(gsrc) : "memory");
#endif
}

__device__ __forceinline__ void wait_async0() {
#if __has_builtin(__builtin_amdgcn_s_wait_asynccnt)
    __builtin_amdgcn_s_wait_asynccnt(0);
#else
    asm volatile("s_wait_asynccnt 0x0" ::: "memory");
#endif
}

// ---------------------------------------------------------------------------
// Kernel 1: per-chunk sums of the rank-1 outer products.
// csum[b][p][c][d] = sum_{t in chunk c} ppm[b,t,p] * out[b,t,d]
// ---------------------------------------------------------------------------
__global__ void __launch_bounds__(512)
chunk_sums(const float* __restrict__ outin,
           const float* __restrict__ ppm,
           float* __restrict__ csum) {
    const int c = blockIdx.x;
    const int b = blockIdx.y;
    const int d = threadIdx.x;

    __shared__ float pch[CH][P_];
    const int tid = threadIdx.x;
    if (tid < CH * P_) {
        const int t = tid / P_, p = tid % P_;
        pch[t][p] = ppm[((size_t)(b * T_ + c * CH + t)) * P_ + p];
    }
    __syncthreads();

    float acc[P_];
#pragma unroll
    for (int p = 0; p < P_; ++p) acc[p] = 0.0f;

#pragma unroll
    for (int t = 0; t < CH; ++t) {
        const float x = outin[((size_t)(b * T_ + c * CH + t)) * D_ + d];
#pragma unroll
        for (int p = 0; p < P_; ++p) acc[p] += pch[t][p] * x;
    }
#pragma unroll
    for (int p = 0; p < P_; ++p)
        csum[(((size_t)(b * P_ + p)) * NC + c) * D_ + d] = acc[p];
}

// ---------------------------------------------------------------------------
// Kernel 2: in-place EXCLUSIVE prefix scan of the chunk sums over c.
// ---------------------------------------------------------------------------
__global__ void __launch_bounds__(512)
chunk_scan(float* __restrict__ csum) {
    const int idx = blockIdx.x * blockDim.x + threadIdx.x;
    if (idx >= B_ * P_ * D_) return;
    const int d  = idx % D_;
    const int bp = idx / D_;
    const size_t base = (size_t)bp * NC * D_ + d;

    float run = 0.0f;
    for (int c = 0; c < NC; ++c) {
        const float v = csum[base + (size_t)c * D_];
        csum[base + (size_t)c * D_] = run;
        run += v;
    }
}

// ---------------------------------------------------------------------------
// Kernel 3: stage the 16x512 out-chunk into LDS with async b128 DMA, compute
// the intra-chunk inclusive prefix via WMMA (L16 @ X), add the inter-chunk
// carry, fused LayerNorm over D, write output.  One block per (b,p,chunk);
// 512 threads = 16 waves; each wave owns two 16-wide d-tiles.
// The LDS buffer xs[][] is reused: staged input during WMMA, then the
// accumulator spill for the LayerNorm row reduction (barrier-separated).
// ---------------------------------------------------------------------------
__global__ void __launch_bounds__(512)
prefix_ln(const float* __restrict__ outin,
          const float* __restrict__ ppm,
          const float* __restrict__ carry,
          const float* __restrict__ a2,
          const float* __restrict__ b2,
          float* __restrict__ y) {
    const int blk = blockIdx.x;
    const int c   = blk % NC;
    const int bp  = blk / NC;
    const int p   = bp % P_;
    const int b   = bp / P_;
    const int t0  = c * CH;

    __shared__ float pch[CH];                                        // ppm chunk
    __shared__ __attribute__((aligned(16))) float xs[CH][D_ + DPAD]; // 33 KB
    __shared__ float s1[CH][NTILE];
    __shared__ float s2[CH][NTILE];
    __shared__ float mrow[CH], irow[CH];

    const int tid  = threadIdx.x;
    const int lane = tid & 31;
    const int wave = tid >> 5;   // 0..15
    const int hi   = lane >> 4;  // 0 = lanes 0-15, 1 = lanes 16-31
    const int lm   = lane & 15;

    // --- Stage out[b, t0..t0+15, 0..511] into xs[t][d] via async DMA. ---
    {
        const int t = tid >> 5;      // row 0..15 (one wave per row)
        const int l = tid & 31;
        const float* grow = outin + ((size_t)(b * T_ + t0 + t)) * D_;
#pragma unroll
        for (int q = 0; q < 4; ++q) {
            const int d0 = (l + 32 * q) * 4;   // 16B packet start (floats)
            async_copy_b128(grow + d0, &xs[t][d0]);
        }
    }
    if (tid < CH)
        pch[tid] = ppm[((size_t)(b * T_ + t0 + tid)) * P_ + p];
    wait_async0();
    __syncthreads();

    v8f acc[2];

#pragma unroll
    for (int i = 0; i < 2; ++i) {
        const int dtile = wave * 2 + i;
        const int dcol  = dtile * 16 + lm;

        v8f cacc = {};
        // P16 = L16 @ X : four chained v_wmma_f32_16x16x4_f32 over K=16.
        // A (16x4 f32): lanes 0-15 -> M=lane, VGPR{0,1}=K{kb+0,kb+1};
        //               lanes 16-31 -> M=lane-16, VGPR{0,1}=K{kb+2,kb+3}.
        // B (4x16 f32): VGPR0 rows {kb+0 (lanes 0-15), kb+2 (lanes 16-31)},
        //               VGPR1 rows {kb+1, kb+3}; N = lane%16.
#pragma unroll
        for (int kk = 0; kk < 4; ++kk) {
            const int kb = kk * 4;
            const int k0 = kb + 2 * hi;

            v2f a;
            a.x = (lm >= k0)     ? 1.0f : 0.0f;   // L[m,k] = (m >= k)
            a.y = (lm >= k0 + 1) ? 1.0f : 0.0f;

            v2f bb;
            bb.x = pch[k0]     * xs[k0][dcol];
            bb.y = pch[k0 + 1] * xs[k0 + 1][dcol];

            cacc = __builtin_amdgcn_wmma_f32_16x16x4_f32(
                false, a, false, bb, (short)0, cacc, false, false);
        }

        // Add inter-chunk carry (broadcast per column d).
        const float cv = carry[(((size_t)(b * P_ + p)) * NC + c) * D_ + dcol];
#pragma unroll
        for (int r = 0; r < 8; ++r) cacc[r] += cv;
        acc[i] = cacc;
    }

    // xs still holds staged input being read above -> barrier before reuse.
    __syncthreads();

    // Spill to row-major LDS: xs[row][d].  VGPR r holds rows r (lanes 0-15)
    // and r+8 (lanes 16-31); column = dtile*16 + lane%16.
#pragma unroll
    for (int i = 0; i < 2; ++i) {
        const int dcol = (wave * 2 + i) * 16 + lm;
#pragma unroll
        for (int r = 0; r < 8; ++r)
            xs[r + 8 * hi][dcol] = acc[i][r];
    }
    __syncthreads();

    // Each thread reduces one (row, 16-d slice): 512 threads = 16 rows x 32 slices.
    {
        const int row = tid & 15;
        const int sl  = tid >> 4;
        const float4* px = (const float4*)&xs[row][sl * 16];
        float S1 = 0.0f, S2 = 0.0f;
#pragma unroll
        for (int j = 0; j < 4; ++j) {
            const float4 v = px[j];
            S1 += v.x + v.y + v.z + v.w;
            S2 += v.x * v.x + v.y * v.y + v.z * v.z + v.w * v.w;
        }
        s1[row][sl] = S1;
        s2[row][sl] = S2;
    }
    __syncthreads();

    // Finish: one thread per row combines the 32 slice partials.
    if (tid < CH) {
        float S1 = 0.0f, S2 = 0.0f;
#pragma unroll
        for (int t = 0; t < NTILE; ++t) { S1 += s1[tid][t]; S2 += s2[tid][t]; }
        const float mean = S1 * (1.0f / D_);
        float var = (S2 - S1 * mean) * (1.0f / (D_ - 1));   // unbiased (ddof=1)
        var = var < 0.0f ? 0.0f : var;
        const float std = sqrtf(var);
        mrow[tid] = mean;
        irow[tid] = 1.0f / (std + 1e-6f);
    }
    __syncthreads();

    // Normalize + affine + store:  y[b, t0+row, p, d]
#pragma unroll
    for (int i = 0; i < 2; ++i) {
        const int dcol = (wave * 2 + i) * 16 + lm;
        const float ga = a2[dcol];
        const float gb = b2[dcol];
#pragma unroll
        for (int r = 0; r < 8; ++r) {
            const int row = r + 8 * hi;
            const float yv = ga * (acc[i][r] - mrow[row]) * irow[row] + gb;
            y[(((size_t)(b * T_ + t0 + row)) * P_ + p) * D_ + dcol] = yv;
        }
    }
}

// ---------------------------------------------------------------------------
extern "C" void kernel_launch(void* const* d_in, const int* in_sizes, int n_in,
                              void* d_out, int out_size, void* d_ws, size_t ws_size,
                              hipStream_t stream) {
    (void)in_sizes; (void)n_in; (void)out_size; (void)ws_size;

    const float* outin = (const float*)d_in[0];  // (B,T,D)
    const float* ppm   = (const float*)d_in[1];  // (B,T,P)
    const float* a2    = (const float*)d_in[2];  // (D,)
    const float* b2    = (const float*)d_in[3];  // (D,)
    float* y    = (float*)d_out;                 // (B,T,P,D)
    float* csum = (float*)d_ws;                  // B*P*NC*D floats = 7 MB

    chunk_sums<<<dim3(NC, B_), 512, 0, stream>>>(outin, ppm, csum);

    const int scan_threads = B_ * P_ * D_;
    chunk_scan<<<(scan_threads + 511) / 512, 512, 0, stream>>>(csum);

    prefix_ln<<<B_ * P_ * NC, 512, 0, stream>>>(outin, ppm, csum, a2, b2, y);
}